// CometBertECTagging_84507776516624
// MI455X (gfx1250) — compile-verified
//
#include <hip/hip_runtime.h>
#include <hip/hip_bf16.h>

// CometBertECTagging fused kernel for gfx1250 (MI455X, wave32, WMMA).
//
// Reference: compaction of bert tokens (cumsum scatter) -> concat with comet
// -> [B*L,1792] x [1792,7] GEMM + bias. Memory-bound (AI ~ 3.5 flop/byte),
// so keep fp32 and use V_WMMA_F32_16X16X4_F32 to do the matmul on the matrix
// pipe while streaming 236 MB at HBM rate.

typedef __attribute__((ext_vector_type(2))) float v2f;
typedef __attribute__((ext_vector_type(8))) float v8f;

#define CB_B   64
#define CB_L   512
#define CB_HB  768
#define CB_HC  1024
#define CB_NL  7
#define CB_K   (CB_HB + CB_HC)     // 1792
#define WSTRIDE (CB_K + 4)         // pad -> half-wave bank-conflict free in LDS

// ---------------------------------------------------------------------------
// Kernel 1: turn the scatter-compaction into a gather map.
// src_idx[b][slot] = source token j (or -1 if that slot is never written).
// ---------------------------------------------------------------------------
__global__ void cb_build_index(const int* __restrict__ valid,
                               int* __restrict__ src_idx) {
    const int b = blockIdx.x;
    const int t = threadIdx.x;                 // 0..511, one thread per token
    __shared__ int v[CB_L];
    v[t] = valid[b * CB_L + t];
    src_idx[b * CB_L + t] = -1;                // default: slot stays zero
    __syncthreads();
    int pos = 0;                               // exclusive prefix of valid bits
    for (int j = 0; j < t; ++j) pos += v[j];
    if (v[t] == 1) src_idx[b * CB_L + pos] = t;
}

// ---------------------------------------------------------------------------
// Kernel 2: fused gather + concat + GEMM(1792x7) + bias via f32 WMMA.
// One wave32 per 16-token tile; N padded 7 -> 16.
//
// V_WMMA_F32_16X16X4_F32 operand layout (wave32):
//   A 16x4 : lane(half,m): m=lane&15 row, half=lane>>4; v0=K(2*half), v1=K(2*half+1)
//   B 4x16 : lane(half,n): n=lane&15 col;               v0=K(2*half), v1=K(2*half+1)
//   C/D    : lane(half,n): vgpr r holds row M = half*8 + r, col N = n
// ---------------------------------------------------------------------------
__global__ void cb_fused_gemm(const float* __restrict__ bert,
                              const float* __restrict__ comet,
                              const int*   __restrict__ src_idx,
                              const float* __restrict__ W,      // [1792][7]
                              const float* __restrict__ bias,   // [7]
                              float*       __restrict__ out) {  // [B*L][7]
    extern __shared__ float wlds[];            // [16][WSTRIDE]

    const int tid  = threadIdx.x;              // 256 threads = 8 waves
    const int lane = tid & 31;
    const int wave = tid >> 5;

    // ---- Stage classifier W transposed + zero-padded into LDS -------------
    // wlds[n][k] = (n < 7) ? W[k*7 + n] : 0
    for (int idx = tid; idx < 16 * CB_K; idx += blockDim.x) {
        const int n = idx & 15;
        const int k = idx >> 4;
        wlds[n * WSTRIDE + k] = (n < CB_NL) ? W[k * CB_NL + n] : 0.0f;
    }
    __syncthreads();

    // ---- Per-lane tile / token setup --------------------------------------
    const int tile = blockIdx.x * 8 + wave;    // 0..2047 (16 tokens each)
    const int m    = lane & 15;                // row within tile / B column
    const int half = lane >> 4;                // K-half select
    const int n    = m;                        // output column for C/D & B
    const int tok  = tile * 16 + m;            // global token
    const int bb   = tok >> 9;                 // / L
    const int li   = tok & (CB_L - 1);         // % L

    const int s        = src_idx[bb * CB_L + li];
    const bool validTk = (s >= 0);
    // Clamp gather pointer so the load is always safe; zero via select.
    const float* bertRow  = bert  + (size_t)(bb * CB_L + (validTk ? s : 0)) * CB_HB;
    const float* cometRow = comet + (size_t)(bb * CB_L + li) * CB_HC;
    const float* wRow     = wlds + n * WSTRIDE;
    const int k0base = 2 * half;

    v8f acc = {};
    const v2f zero2 = {0.0f, 0.0f};

    // ---- K loop, bert (gathered/compacted) region: k in [0,768) -----------
#pragma unroll 4
    for (int kk = 0; kk < CB_HB; kk += 4) {
        const int k0 = kk + k0base;
        v2f av = *(const v2f*)(bertRow + k0);
        v2f a  = validTk ? av : zero2;         // select keeps EXEC all-1s
        v2f bv = *(const v2f*)(wRow + k0);
        acc = __builtin_amdgcn_wmma_f32_16x16x4_f32(
            /*neg_a=*/false, a, /*neg_b=*/false, bv,
            /*c_mod=*/(short)0, acc, /*reuse_a=*/false, /*reuse_b=*/false);
    }
    // ---- K loop, comet region: k in [768,1792) ----------------------------
#pragma unroll 4
    for (int kk = CB_HB; kk < CB_K; kk += 4) {
        const int k0 = kk + k0base;
        v2f a  = *(const v2f*)(cometRow + (k0 - CB_HB));
        v2f bv = *(const v2f*)(wRow + k0);
        acc = __builtin_amdgcn_wmma_f32_16x16x4_f32(
            false, a, false, bv, (short)0, acc, false, false);
    }

    // ---- Bias + store (only N < 7 is real output) -------------------------
    if (n < CB_NL) {
        const float bn = bias[n];
        const int rowBase = tile * 16 + half * 8;
#pragma unroll
        for (int r = 0; r < 8; ++r) {
            out[(size_t)(rowBase + r) * CB_NL + n] = acc[r] + bn;
        }
    }
}

// ---------------------------------------------------------------------------
extern "C" void kernel_launch(void* const* d_in, const int* in_sizes, int n_in,
                              void* d_out, int out_size, void* d_ws, size_t ws_size,
                              hipStream_t stream) {
    const float* bert  = (const float*)d_in[0];
    const float* comet = (const float*)d_in[1];
    const int*   valid = (const int*)d_in[2];
    const float* W     = (const float*)d_in[3];
    const float* bias  = (const float*)d_in[4];
    float* out = (float*)d_out;
    int* src_idx = (int*)d_ws;                 // B*L ints = 128 KB scratch

    cb_build_index<<<CB_B, CB_L, 0, stream>>>(valid, src_idx);

    const size_t ldsBytes = (size_t)16 * WSTRIDE * sizeof(float); // ~112 KB
    // 2048 tiles of 16 tokens; 8 waves (256 threads) per block -> 256 blocks
    cb_fused_gemm<<<256, 256, ldsBytes, stream>>>(bert, comet, src_idx, W, bias, out);
}